// GATFuse_45638322487544
// MI455X (gfx1250) — compile-verified
//
#include <hip/hip_runtime.h>
#include <math.h>

#define NN 50000
#define EE 800000
#define GG 512
#define INDIM 128
#define NEGSLOPE 0.2f
#define BNEPS 1e-5f

typedef float v2f __attribute__((ext_vector_type(2)));
typedef float v8f __attribute__((ext_vector_type(8)));

static inline int cdiv_i(long long a, long long b) { return (int)((a + b - 1) / b); }

// ---------------- utility kernels ----------------

__global__ void k_fill(float* __restrict__ p, float v, long long n) {
  long long i = (long long)blockIdx.x * blockDim.x + threadIdx.x;
  if (i < n) p[i] = v;
}

__global__ void k_dup(const float* __restrict__ x, float* __restrict__ a,
                      float* __restrict__ b, long long n) {
  long long i = (long long)blockIdx.x * blockDim.x + threadIdx.x;
  if (i < n) { float v = x[i]; a[i] = v; b[i] = v; }
}

// ---------------- WMMA GEMM: Hout[nrows x ncols] = X[nrows x 128] * W[128 x ncols] ----------------
// 256 threads = 8 waves; each wave owns one 16x16 C tile. W staged in LDS.
// A frag (16x4 f32): lane m=lane&15, k-pair base kb=2*(lane>>4), vgprs hold k=kb,kb+1.
// B frag (4x16 f32): lane n=lane&15, same kb split.
// C (16x16 f32): vgpr r -> row = 8*(lane>>4)+r, col = lane&15.
__global__ void __launch_bounds__(256)
k_gemm_wmma(const float* __restrict__ X, const float* __restrict__ W,
            float* __restrict__ Hout, int nrows, int ncols) {
  __shared__ float Wl[INDIM * INDIM];
  const int tid = threadIdx.x;
  const int tot = INDIM * ncols;
  for (int i = tid * 4; i < tot; i += 256 * 4)
    *(float4*)&Wl[i] = *(const float4*)&W[i];
  __syncthreads();

  const int wave = tid >> 5, lane = tid & 31;
  const int colTiles = ncols >> 4;          // 8 (HC=128) or 4 (HC=64)
  const int rtpb = 8 / colTiles;            // row tiles per block
  const int rowTile = blockIdx.x * rtpb + wave / colTiles;
  if (rowTile * 16 >= nrows) return;        // wave-uniform: EXEC stays all-1s
  const int colBase = (wave % colTiles) << 4;
  const int ln = lane & 15;
  const int kb = (lane >> 4) << 1;

  const float* xrow = X + (long long)(rowTile * 16 + ln) * INDIM;
  v8f acc = {};
#pragma unroll
  for (int k = 0; k < INDIM; k += 4) {
    v2f a, b;
    a.x = xrow[k + kb];
    a.y = xrow[k + kb + 1];
    b.x = Wl[(k + kb) * ncols + colBase + ln];
    b.y = Wl[(k + kb + 1) * ncols + colBase + ln];
    acc = __builtin_amdgcn_wmma_f32_16x16x4_f32(false, a, false, b,
                                                (short)0, acc, false, false);
  }
  const int row0 = rowTile * 16 + ((lane >> 4) << 3);
  float* orow = Hout + (long long)row0 * ncols + colBase + ln;
#pragma unroll
  for (int r = 0; r < 8; ++r)
    orow[(long long)r * ncols] = acc[r];
}

// ---------------- attention logits: als/ald [N, heads] ----------------
// wave per node; lane covers 4 channels (C=64 -> 16 lanes per head).
__global__ void k_att_logits(const float* __restrict__ H, const float* __restrict__ asrc,
                             const float* __restrict__ adst, float* __restrict__ als,
                             float* __restrict__ ald, int N, int heads) {
  const int node = blockIdx.x * (blockDim.x >> 5) + (threadIdx.x >> 5);
  if (node >= N) return;
  const int lane = threadIdx.x & 31;
  const int HC = heads * 64;
  float ss = 0.f, sd = 0.f;
  const int c0 = lane * 4;
  if (c0 < HC) {
    const float4 hv = *(const float4*)(H + (long long)node * HC + c0);
    const float4 as = *(const float4*)(asrc + c0);
    const float4 ad = *(const float4*)(adst + c0);
    ss = hv.x * as.x + hv.y * as.y + hv.z * as.z + hv.w * as.w;
    sd = hv.x * ad.x + hv.y * ad.y + hv.z * ad.z + hv.w * ad.w;
  }
#pragma unroll
  for (int off = 1; off < 16; off <<= 1) {    // reduce within 16-lane head group
    ss += __shfl_xor(ss, off, 32);
    sd += __shfl_xor(sd, off, 32);
  }
  if ((lane & 15) == 0) {
    const int h = lane >> 4;
    if (h < heads) {
      als[(long long)node * heads + h] = ss;
      ald[(long long)node * heads + h] = sd;
    }
  }
}

// ---------------- edge passes ----------------

__device__ __forceinline__ float leakyr(float v) { return v > 0.f ? v : NEGSLOPE * v; }

__device__ __forceinline__ void atomicMaxF(float* addr, float v) {
  if (v >= 0.f) atomicMax((int*)addr, __float_as_int(v));
  else          atomicMin((unsigned int*)addr, __float_as_uint(v));
}

__global__ void k_edge_max(const float* __restrict__ als, const float* __restrict__ ald,
                           const int* __restrict__ src, const int* __restrict__ dst,
                           float* __restrict__ m, int E, int N, int heads) {
  const int i = blockIdx.x * blockDim.x + threadIdx.x;
  if (i >= E + N) return;
  const int s = (i < E) ? src[i] : (i - E);   // self-loops appended
  const int d = (i < E) ? dst[i] : (i - E);
  for (int h = 0; h < heads; ++h) {
    float al = leakyr(als[(long long)s * heads + h] + ald[(long long)d * heads + h]);
    atomicMaxF(&m[(long long)d * heads + h], al);
  }
}

__global__ void k_fix_m(float* __restrict__ m, int n) {
  const int i = blockIdx.x * blockDim.x + threadIdx.x;
  if (i >= n) return;
  float v = m[i];
  if (!(v > -1e37f) || v > 1e37f || v != v) m[i] = 0.f;
}

__global__ void k_edge_exp(const float* __restrict__ als, const float* __restrict__ ald,
                           const int* __restrict__ src, const int* __restrict__ dst,
                           const float* __restrict__ m, float* __restrict__ e,
                           float* __restrict__ denom, int E, int N, int heads) {
  const int i = blockIdx.x * blockDim.x + threadIdx.x;
  if (i >= E + N) return;
  const int s = (i < E) ? src[i] : (i - E);
  const int d = (i < E) ? dst[i] : (i - E);
  for (int h = 0; h < heads; ++h) {
    float al = leakyr(als[(long long)s * heads + h] + ald[(long long)d * heads + h]);
    float ev = expf(al - m[(long long)d * heads + h]);
    e[(long long)i * heads + h] = ev;
    atomicAdd(&denom[(long long)d * heads + h], ev);
  }
}

// flat over (E+N)*HC: coalesced gather of h[src], atomic scatter into out[dst]
__global__ void k_edge_scatter(const float* __restrict__ H, const int* __restrict__ src,
                               const int* __restrict__ dst, const float* __restrict__ e,
                               const float* __restrict__ denom, float* __restrict__ out,
                               int E, int N, int heads, int HC) {
  const long long idx = (long long)blockIdx.x * blockDim.x + threadIdx.x;
  const long long total = (long long)(E + N) * HC;
  if (idx >= total) return;
  const int c = (int)(idx % HC);
  const int i = (int)(idx / HC);
  const int s = (i < E) ? src[i] : (i - E);
  const int d = (i < E) ? dst[i] : (i - E);
  const int h = c >> 6;                        // C = 64
  const float coef = e[(long long)i * heads + h] /
                     (denom[(long long)d * heads + h] + 1e-16f);
  atomicAdd(out + (long long)d * HC + c, coef * H[(long long)s * HC + c]);
}

// ---------------- batch norm (train mode) + relu ----------------
// blockDim.x == HC; each thread owns a channel, strides over rows.
__global__ void k_col_stats(const float* __restrict__ X, float* __restrict__ sum,
                            float* __restrict__ sumsq, int N, int HC) {
  const int c = threadIdx.x;
  float s = 0.f, s2 = 0.f;
  for (int r = blockIdx.x; r < N; r += gridDim.x) {
    float v = X[(long long)r * HC + c];
    s += v; s2 += v * v;
  }
  atomicAdd(&sum[c], s);
  atomicAdd(&sumsq[c], s2);
}

__global__ void k_bn_relu(const float* __restrict__ X, const float* __restrict__ sum,
                          const float* __restrict__ sumsq, const float* __restrict__ g,
                          const float* __restrict__ b, float* __restrict__ Y,
                          int N, int HC) {
  const long long idx = (long long)blockIdx.x * blockDim.x + threadIdx.x;
  if (idx >= (long long)N * HC) return;
  const int c = (int)(idx % HC);
  const float mu = sum[c] / (float)N;
  const float var = sumsq[c] / (float)N - mu * mu;  // biased var
  float v = (X[idx] - mu) * rsqrtf(var + BNEPS) * g[c] + b[c];
  Y[idx] = v > 0.f ? v : 0.f;
}

// ---------------- final: xf=(f0+f1)/2+bf ; v = xf.Wo + bo ; scatter by graph ----------------
__global__ void k_final(const float* __restrict__ f0, const float* __restrict__ f1,
                        const float* __restrict__ bf, const float* __restrict__ Wo,
                        const float* __restrict__ bo, const int* __restrict__ batch,
                        float* __restrict__ gsum, float* __restrict__ gcnt, int N) {
  const int node = blockIdx.x * (blockDim.x >> 5) + (threadIdx.x >> 5);
  if (node >= N) return;
  const int lane = threadIdx.x & 31;
  float acc = 0.f;
#pragma unroll
  for (int j = 0; j < 2; ++j) {               // 64 channels, 2 per lane
    const int c = lane * 2 + j;
    float xf = (f0[(long long)node * 64 + c] + f1[(long long)node * 64 + c]) * 0.5f + bf[c];
    acc += xf * Wo[c];
  }
#pragma unroll
  for (int off = 1; off < 32; off <<= 1) acc += __shfl_xor(acc, off, 32);
  if (lane == 0) {
    const int gidx = batch[node];
    atomicAdd(&gsum[gidx], acc + bo[0]);
    atomicAdd(&gcnt[gidx], 1.0f);
  }
}

__global__ void k_divout(const float* __restrict__ gsum, const float* __restrict__ gcnt,
                         float* __restrict__ out, int G) {
  const int g = blockIdx.x * blockDim.x + threadIdx.x;
  if (g < G) out[g] = gsum[g] / fmaxf(gcnt[g], 1.0f);
}

// ---------------- one gat_conv (no bias; bias either cancels in BN or is added in k_final) ----------------
static void run_gat(const float* xin, const float* W, const float* asrc, const float* adst,
                    const int* ei, int heads, float* hbuf, float* outbuf,
                    float* als, float* ald, float* mbuf, float* dn, float* ebuf,
                    hipStream_t stream) {
  const int HC = heads * 64;
  const int rtpb = 8 / (HC >> 4);
  const int rowTiles = NN / 16;                 // 3125, exact
  k_gemm_wmma<<<cdiv_i(rowTiles, rtpb), 256, 0, stream>>>(xin, W, hbuf, NN, HC);
  k_att_logits<<<cdiv_i(NN, 8), 256, 0, stream>>>(hbuf, asrc, adst, als, ald, NN, heads);

  const long long nmh = (long long)NN * heads;
  k_fill<<<cdiv_i(nmh, 256), 256, 0, stream>>>(mbuf, -INFINITY, nmh);
  k_fill<<<cdiv_i(nmh, 256), 256, 0, stream>>>(dn, 0.f, nmh);
  k_fill<<<cdiv_i((long long)NN * HC, 256), 256, 0, stream>>>(outbuf, 0.f, (long long)NN * HC);

  const int* src = ei;
  const int* dst = ei + EE;
  const int Etot = EE + NN;
  k_edge_max<<<cdiv_i(Etot, 256), 256, 0, stream>>>(als, ald, src, dst, mbuf, EE, NN, heads);
  k_fix_m<<<cdiv_i(nmh, 256), 256, 0, stream>>>(mbuf, (int)nmh);
  k_edge_exp<<<cdiv_i(Etot, 256), 256, 0, stream>>>(als, ald, src, dst, mbuf, ebuf, dn, EE, NN, heads);
  const long long tot = (long long)Etot * HC;
  k_edge_scatter<<<cdiv_i(tot, 256), 256, 0, stream>>>(hbuf, src, dst, ebuf, dn, outbuf,
                                                       EE, NN, heads, HC);
}

extern "C" void kernel_launch(void* const* d_in, const int* in_sizes, int n_in,
                              void* d_out, int out_size, void* d_ws, size_t ws_size,
                              hipStream_t stream) {
  const float* x       = (const float*)d_in[0];
  const int*   ei_sc   = (const int*)  d_in[1];
  const int*   ei_fc   = (const int*)  d_in[2];
  const int*   batch   = (const int*)  d_in[3];
  const float* Ws      = (const float*)d_in[4];
  const float* att_src = (const float*)d_in[5];
  const float* att_dst = (const float*)d_in[6];
  // d_in[7] = bgat: constant shift cancels exactly inside train-mode BatchNorm
  const float* bn_g    = (const float*)d_in[8];
  const float* bn_b    = (const float*)d_in[9];
  const float* Wf      = (const float*)d_in[10];
  const float* af_src  = (const float*)d_in[11];
  const float* af_dst  = (const float*)d_in[12];
  const float* bf      = (const float*)d_in[13];
  const float* Wo      = (const float*)d_in[14];
  const float* bo      = (const float*)d_in[15];
  float* outp = (float*)d_out;

  // ---- workspace layout (floats) ----
  float* ws = (float*)d_ws;
  const long long NH = (long long)NN * INDIM;        // 6.4M
  float* x0   = ws;                                  // N*128
  float* x1   = x0 + NH;                             // N*128
  float* hb   = x1 + NH;                             // N*128
  float* ob   = hb + NH;                             // N*128 (also fin0|fin1 for final layer)
  float* als  = ob + NH;                             // N*2
  float* ald  = als + (long long)NN * 2;             // N*2
  float* mbuf = ald + (long long)NN * 2;             // N*2
  float* dn   = mbuf + (long long)NN * 2;            // N*2
  float* ebuf = dn + (long long)NN * 2;              // (E+N)*2
  float* csum = ebuf + (long long)(EE + NN) * 2;     // 128
  float* csq  = csum + INDIM;                        // 128
  float* gsum = csq + INDIM;                         // 512
  float* gcnt = gsum + GG;                           // 512

  k_dup<<<cdiv_i(NH, 256), 256, 0, stream>>>(x, x0, x1, NH);

  float*      xb[2]  = { x0, x1 };
  const int*  eis[2] = { ei_sc, ei_fc };

  for (int l = 0; l < 3; ++l) {
    for (int b = 0; b < 2; ++b) {
      run_gat(xb[b], Ws + (long long)l * INDIM * INDIM,
              att_src + (long long)l * 128, att_dst + (long long)l * 128,
              eis[b], /*heads=*/2, hb, ob, als, ald, mbuf, dn, ebuf, stream);
      k_fill<<<1, 256, 0, stream>>>(csum, 0.f, 256);        // csum + csq contiguous
      k_col_stats<<<256, INDIM, 0, stream>>>(ob, csum, csq, NN, INDIM);
      k_bn_relu<<<cdiv_i(NH, 256), 256, 0, stream>>>(ob, csum, csq,
                                                     bn_g + (long long)l * INDIM,
                                                     bn_b + (long long)l * INDIM,
                                                     xb[b], NN, INDIM);
    }
  }

  // final single-head GAT per branch; outputs packed into the two halves of ob
  float* fin0 = ob;
  float* fin1 = ob + (long long)NN * 64;
  run_gat(x0, Wf, af_src, af_dst, ei_sc, /*heads=*/1, hb, fin0, als, ald, mbuf, dn, ebuf, stream);
  run_gat(x1, Wf, af_src, af_dst, ei_fc, /*heads=*/1, hb, fin1, als, ald, mbuf, dn, ebuf, stream);

  k_fill<<<cdiv_i(2 * GG, 256), 256, 0, stream>>>(gsum, 0.f, 2 * GG);  // gsum + gcnt contiguous
  k_final<<<cdiv_i(NN, 8), 256, 0, stream>>>(fin0, fin1, bf, Wo, bo, batch, gsum, gcnt, NN);
  k_divout<<<cdiv_i(GG, 256), 256, 0, stream>>>(gsum, gcnt, outp, GG);
}